// ASAP_58033598104024
// MI455X (gfx1250) — compile-verified
//
#include <hip/hip_runtime.h>

#define G      8
#define NPAD   128     // N_PER_G (padded node count per graph)
#define HID    64
#define FIN    128
#define NL     10
#define NCLS   10
#define SLOPE  0.2f

typedef float v2f __attribute__((ext_vector_type(2)));
typedef float v4f __attribute__((ext_vector_type(4)));
typedef float v8f __attribute__((ext_vector_type(8)));

// ---------------------------------------------------------------------------
// f32 WMMA step: C(16x16) += A(16x4) * B(4x16), native fp32 matrix pipe.
// ---------------------------------------------------------------------------
__device__ __forceinline__ v8f wstep(v8f c, v2f a, v2f b) {
  return __builtin_amdgcn_wmma_f32_16x16x4_f32(false, a, false, b,
                                               (short)0, c, false, false);
}
__device__ __forceinline__ v8f vzero8() {
  v8f z = {0.f,0.f,0.f,0.f,0.f,0.f,0.f,0.f};
  return z;
}

// ---------------------------------------------------------------------------
// Utility kernels
// ---------------------------------------------------------------------------
__global__ void kZero(float* p, int n) {
  for (int i = blockIdx.x * blockDim.x + threadIdx.x; i < n;
       i += gridDim.x * blockDim.x)
    p[i] = 0.0f;
}

__global__ void kAdj(const int* ei, int E, float* D) {
  int e = blockIdx.x * blockDim.x + threadIdx.x;
  if (e >= E) return;
  int s = ei[e], d = ei[E + e];
  int g = s / NPAD;
  int si = s - g * NPAD;
  int dj = d - g * NPAD;
  D[(size_t)g * NPAD * NPAD + si * NPAD + dj] = 1.0f;   // .set(1.0), races benign
}

// stores 1 / max(colnnz, 1)  (reciprocal so the GEMM epilogue is a v_mul)
__global__ void kDeg(const float* D, float* rdeg) {
  int idx = blockIdx.x * blockDim.x + threadIdx.x;
  if (idx >= G * NPAD) return;
  int g = idx >> 7, j = idx & 127;
  const float* Dg = D + (size_t)g * NPAD * NPAD;
  int c = 0;
  for (int i = 0; i < NPAD; i++) c += (Dg[i * NPAD + j] > 0.0f) ? 1 : 0;
  rdeg[idx] = 1.0f / (float)(c > 1 ? c : 1);
}

// ---------------------------------------------------------------------------
// agg = (D^T @ X) * rdeg[row] — WMMA, one wave computes a 16x64 strip
// grid.x = 8 * (F/64), grid.y = G
// ---------------------------------------------------------------------------
__global__ void kAggT(const float* D, const float* X, const float* rdeg,
                      float* OUT, int F) {
  int g  = blockIdx.y;
  int ng = F >> 6;                      // 64-col groups
  int m0 = (blockIdx.x / ng) << 4;
  int nb = (blockIdx.x % ng) << 6;
  const float* Dg = D + (size_t)g * NPAD * NPAD;
  const float* Xg = X + (size_t)g * NPAD * F;
  int lane = threadIdx.x & 31;
  int row  = lane & 15;
  int kb   = (lane >> 4) << 1;          // 0 or 2
  v8f acc[4];
#pragma unroll
  for (int nt = 0; nt < 4; nt++) acc[nt] = vzero8();
  for (int k0 = 0; k0 < NPAD; k0 += 4) {
    v2f a;                               // A[m][k] = D[k][m] (transposed, strided)
    a.x = Dg[(k0 + kb + 0) * NPAD + m0 + row];
    a.y = Dg[(k0 + kb + 1) * NPAD + m0 + row];
#pragma unroll
    for (int nt = 0; nt < 4; nt++) {
      int col = nb + (nt << 4) + row;
      v2f b;
      b.x = Xg[(k0 + kb + 0) * F + col];
      b.y = Xg[(k0 + kb + 1) * F + col];
      acc[nt] = wstep(acc[nt], a, b);
    }
  }
  int half = lane >> 4;
  // 8 contiguous reciprocal-degree values for this lane's rows: two b128 loads,
  // hoisted so the store loop has no load->store round-trips.
  const float* rdp = rdeg + g * NPAD + m0 + half * 8;
  v4f r0 = *(const v4f*)(rdp + 0);
  v4f r1 = *(const v4f*)(rdp + 4);
  float rv[8] = {r0.x, r0.y, r0.z, r0.w, r1.x, r1.y, r1.z, r1.w};
  float* Og = OUT + (size_t)g * NPAD * F;
#pragma unroll
  for (int nt = 0; nt < 4; nt++)
    for (int r = 0; r < 8; r++) {
      int mm = m0 + r + half * 8;
      Og[mm * F + nb + (nt << 4) + row] = acc[nt][r] * rv[r];
    }
}

// ---------------------------------------------------------------------------
// H = relu(agg @ Wr^T + X @ Wt^T + b), rows >= n zeroed — WMMA 16x64 strip
// grid.x = 8, grid.y = G
// ---------------------------------------------------------------------------
__global__ void kConvOut(const float* AGGm, const float* X, const float* Wr,
                         const float* bias, const float* Wt, float* H,
                         int F, int n) {
  int g  = blockIdx.y;
  int m0 = blockIdx.x << 4;
  const float* Ag = AGGm + (size_t)g * NPAD * F;
  const float* Xg = X + (size_t)g * NPAD * F;
  int lane = threadIdx.x & 31;
  int row  = lane & 15;
  int kb   = (lane >> 4) << 1;
  v8f acc[4];
#pragma unroll
  for (int nt = 0; nt < 4; nt++) acc[nt] = vzero8();
  for (int k0 = 0; k0 < F; k0 += 4) {
    v2f a = *(const v2f*)(Ag + (m0 + row) * F + k0 + kb);   // b64 load
#pragma unroll
    for (int nt = 0; nt < 4; nt++) {
      v2f b = *(const v2f*)(Wr + ((nt << 4) + row) * F + k0 + kb);
      acc[nt] = wstep(acc[nt], a, b);
    }
  }
  for (int k0 = 0; k0 < F; k0 += 4) {
    v2f a = *(const v2f*)(Xg + (m0 + row) * F + k0 + kb);
#pragma unroll
    for (int nt = 0; nt < 4; nt++) {
      v2f b = *(const v2f*)(Wt + ((nt << 4) + row) * F + k0 + kb);
      acc[nt] = wstep(acc[nt], a, b);
    }
  }
  int half = lane >> 4;
  float* Hg = H + (size_t)g * NPAD * HID;
#pragma unroll
  for (int nt = 0; nt < 4; nt++) {
    int col = (nt << 4) + row;
    float bs = bias[col];
    for (int r = 0; r < 8; r++) {
      int mm = m0 + r + half * 8;
      float v = acc[nt][r] + bs;
      v = v > 0.0f ? v : 0.0f;
      Hg[mm * HID + col] = (mm < n) ? v : 0.0f;
    }
  }
}

__global__ void kMean(const float* H, float* xsl, int n) {
  int g = blockIdx.x, c = threadIdx.x;
  const float* Hg = H + (size_t)g * NPAD * HID;
  float s = 0.0f;
  for (int i = 0; i < n; i++) s += Hg[i * HID + c];
  xsl[g * HID + c] = s / (float)n;
}

// ---------------------------------------------------------------------------
// Pooling helpers
// ---------------------------------------------------------------------------
__global__ void kDsl(const float* D, float* DSL) {
  int idx = blockIdx.x * blockDim.x + threadIdx.x;
  if (idx >= G * NPAD * NPAD) return;
  int r  = (idx >> 7) & 127;
  int cc = idx & 127;
  float v = D[idx];
  if (r == cc && v == 0.0f) v = 1.0f;   // add self loop where diag==0
  DSL[idx] = v;
}

__global__ void kDegf(const float* DSL, float* degf) {
  int idx = blockIdx.x * blockDim.x + threadIdx.x;
  if (idx >= G * NPAD) return;
  int g = idx >> 7, j = idx & 127;
  const float* Dg = DSL + (size_t)g * NPAD * NPAD;
  int c = 0;
  for (int i = 0; i < NPAD; i++) c += (Dg[i * NPAD + j] > 0.0f) ? 1 : 0;
  degf[idx] = (float)c;
}

__global__ void kXq(const float* DSL, const float* H, float* XQ) {
  int idx = blockIdx.x * blockDim.x + threadIdx.x;
  if (idx >= G * NPAD * HID) return;
  int g  = idx / (NPAD * HID);
  int r  = idx % (NPAD * HID);
  int j  = r / HID, ch = r % HID;
  const float* Dg = DSL + (size_t)g * NPAD * NPAD;
  const float* Hg = H + (size_t)g * NPAD * HID;
  float m = -3.4e38f;
  for (int i = 0; i < NPAD; i++)
    if (Dg[i * NPAD + j] > 0.0f) {
      float v = Hg[i * HID + ch];
      m = v > m ? v : m;
    }
  XQ[idx] = m;
}

// xq_lin = XQ @ Wl^T + bl — WMMA 16x64 strip, grid (8, G)
__global__ void kXqLin(const float* XQ, const float* Wl, const float* bl,
                       float* XQL) {
  int g  = blockIdx.y;
  int m0 = blockIdx.x << 4;
  const float* Ag = XQ + (size_t)g * NPAD * HID;
  int lane = threadIdx.x & 31;
  int row  = lane & 15;
  int kb   = (lane >> 4) << 1;
  v8f acc[4];
#pragma unroll
  for (int nt = 0; nt < 4; nt++) acc[nt] = vzero8();
  for (int k0 = 0; k0 < HID; k0 += 4) {
    v2f a = *(const v2f*)(Ag + (m0 + row) * HID + k0 + kb);
#pragma unroll
    for (int nt = 0; nt < 4; nt++) {
      v2f b = *(const v2f*)(Wl + ((nt << 4) + row) * HID + k0 + kb);
      acc[nt] = wstep(acc[nt], a, b);
    }
  }
  int half = lane >> 4;
  float* Og = XQL + (size_t)g * NPAD * HID;
#pragma unroll
  for (int nt = 0; nt < 4; nt++) {
    int col = (nt << 4) + row;
    float bs = bl[col];
    for (int r = 0; r < 8; r++) {
      int mm = m0 + r + half * 8;
      Og[mm * HID + col] = acc[nt][r] + bs;
    }
  }
}

__global__ void kVecs(const float* H, const float* XQL, const float* Wb,
                      const float* Wa, float* XB, float* QA) {
  int idx = blockIdx.x * blockDim.x + threadIdx.x;
  if (idx >= G * NPAD) return;
  int g = idx >> 7, i = idx & 127;
  const float* hr = H + (size_t)g * NPAD * HID + i * HID;
  const float* qr = XQL + (size_t)g * NPAD * HID + i * HID;
  float sb = 0.0f, sa = 0.0f;
  for (int c = 0; c < HID; c++) { sb += hr[c] * Wb[c]; sa += qr[c] * Wa[c]; }
  XB[idx] = sb;
  QA[idx] = sa;
}

// per-column masked leaky-relu softmax over rows
__global__ void kScore(const float* DSL, const float* XB, const float* QA,
                       const float* baP, float* S) {
  int idx = blockIdx.x * blockDim.x + threadIdx.x;
  if (idx >= G * NPAD) return;
  int g = idx >> 7, j = idx & 127;
  const float* Dg = DSL + (size_t)g * NPAD * NPAD;
  const float* xb = XB + g * NPAD;
  const float* qa = QA + g * NPAD;
  float* Sg = S + (size_t)g * NPAD * NPAD;
  float ba = baP[0];
  float qj = qa[j] + ba;
  float mx = -3.4e38f;
  for (int i = 0; i < NPAD; i++)
    if (Dg[i * NPAD + j] > 0.0f) {
      float s = xb[i] + qj;
      s = s < 0.0f ? SLOPE * s : s;
      mx = s > mx ? s : mx;
    }
  float sum = 0.0f;
  for (int i = 0; i < NPAD; i++) {
    float e = 0.0f;
    if (Dg[i * NPAD + j] > 0.0f) {
      float s = xb[i] + qj;
      s = s < 0.0f ? SLOPE * s : s;
      e = expf(s - mx);
      sum += e;
    }
    Sg[i * NPAD + j] = e;
  }
  float inv = 1.0f / sum;
  for (int i = 0; i < NPAD; i++) Sg[i * NPAD + j] *= inv;
}

// x_new = S^T @ H — WMMA 16x64 strip, grid (8, G)
__global__ void kXnew(const float* S, const float* H, float* XN) {
  int g  = blockIdx.y;
  int m0 = blockIdx.x << 4;
  const float* Sg = S + (size_t)g * NPAD * NPAD;
  const float* Hg = H + (size_t)g * NPAD * HID;
  int lane = threadIdx.x & 31;
  int row  = lane & 15;
  int kb   = (lane >> 4) << 1;
  v8f acc[4];
#pragma unroll
  for (int nt = 0; nt < 4; nt++) acc[nt] = vzero8();
  for (int k0 = 0; k0 < NPAD; k0 += 4) {
    v2f a;                               // S^T access (strided)
    a.x = Sg[(k0 + kb + 0) * NPAD + m0 + row];
    a.y = Sg[(k0 + kb + 1) * NPAD + m0 + row];
#pragma unroll
    for (int nt = 0; nt < 4; nt++) {
      int col = (nt << 4) + row;
      v2f b;
      b.x = Hg[(k0 + kb + 0) * HID + col];
      b.y = Hg[(k0 + kb + 1) * HID + col];
      acc[nt] = wstep(acc[nt], a, b);
    }
  }
  int half = lane >> 4;
  float* Og = XN + (size_t)g * NPAD * HID;
#pragma unroll
  for (int nt = 0; nt < 4; nt++)
    for (int r = 0; r < 8; r++) {
      int mm = m0 + r + half * 8;
      Og[mm * HID + (nt << 4) + row] = acc[nt][r];
    }
}

__global__ void kAB(const float* XN, const float* W1, const float* b1,
                    const float* W2, const float* W3, const float* b3,
                    float* SA, float* SB, float* SC) {
  int idx = blockIdx.x * blockDim.x + threadIdx.x;
  if (idx >= G * NPAD) return;
  int g = idx >> 7, i = idx & 127;
  const float* xr = XN + (size_t)g * NPAD * HID + i * HID;
  float a = 0.0f, b = 0.0f, c3 = 0.0f;
  for (int c = 0; c < HID; c++) {
    a  += xr[c] * W1[c];
    b  += xr[c] * W2[c];
    c3 += xr[c] * W3[c];
  }
  SA[idx] = a + b1[0];
  SB[idx] = b;
  SC[idx] = c3 + b3[0];
}

__global__ void kFit(const float* DSL, const float* DEGF, const float* SA,
                     const float* SB, const float* SC, float* FIT) {
  int idx = blockIdx.x * blockDim.x + threadIdx.x;
  if (idx >= G * NPAD) return;
  int g = idx >> 7, j = idx & 127;
  const float* Dg = DSL + (size_t)g * NPAD * NPAD;
  const float* sb = SB + g * NPAD;
  float sumb = 0.0f;
  for (int i = 0; i < NPAD; i++)
    if (Dg[i * NPAD + j] > 0.0f) sumb += sb[i];
  float t = DEGF[idx] * SA[idx] - sumb + SC[idx];
  FIT[idx] = 1.0f / (1.0f + expf(-t));
}

// exact top-k (descending, ties -> lowest index) — matches jax.lax.top_k
__global__ void kTopK(const float* FIT, int n, int k, int* PERM) {
  int g = blockIdx.x * blockDim.x + threadIdx.x;
  if (g >= G) return;
  float f[NPAD];
  for (int j = 0; j < n; j++) f[j] = FIT[g * NPAD + j];
  for (int r = 0; r < k; r++) {
    float best = -3.4e38f;
    int bi = 0;
    for (int j = 0; j < n; j++)
      if (f[j] > best) { best = f[j]; bi = j; }
    PERM[g * NPAD + r] = bi;
    f[bi] = -3.4e38f;
  }
}

__global__ void kXout(const float* XN, const float* FIT, const int* PERM,
                      int k, float* HO) {
  int idx = blockIdx.x * blockDim.x + threadIdx.x;
  if (idx >= G * NPAD * HID) return;
  int g = idx / (NPAD * HID);
  int r = (idx % (NPAD * HID)) / HID;
  int ch = idx % HID;
  float v = 0.0f;
  if (r < k) {
    int p = PERM[g * NPAD + r];
    v = XN[(size_t)g * NPAD * HID + p * HID + ch] * FIT[g * NPAD + p];
  }
  HO[idx] = v;
}

// T = D_sl @ S_sel — WMMA 16x64 strip; gathered B columns, branchless mask
// grid.x = 8 * 2, grid.y = G
__global__ void kT(const float* DSL, const float* S, const int* PERM, int k,
                   float* T) {
  int g  = blockIdx.y;
  int m0 = (blockIdx.x >> 1) << 4;
  int nb = (blockIdx.x & 1) << 6;
  const float* Dg = DSL + (size_t)g * NPAD * NPAD;
  const float* Sg = S + (size_t)g * NPAD * NPAD;
  const int* pg = PERM + g * NPAD;
  int lane = threadIdx.x & 31;
  int row  = lane & 15;
  int kb   = (lane >> 4) << 1;
  int pc[4];
  float cm[4];
#pragma unroll
  for (int nt = 0; nt < 4; nt++) {
    int col = nb + (nt << 4) + row;
    pc[nt] = (col < k) ? pg[col] : 0;
    cm[nt] = (col < k) ? 1.0f : 0.0f;
  }
  v8f acc[4];
#pragma unroll
  for (int nt = 0; nt < 4; nt++) acc[nt] = vzero8();
  for (int k0 = 0; k0 < NPAD; k0 += 4) {
    v2f a = *(const v2f*)(Dg + (m0 + row) * NPAD + k0 + kb);
#pragma unroll
    for (int nt = 0; nt < 4; nt++) {
      v2f b;
      b.x = Sg[(k0 + kb + 0) * NPAD + pc[nt]] * cm[nt];
      b.y = Sg[(k0 + kb + 1) * NPAD + pc[nt]] * cm[nt];
      acc[nt] = wstep(acc[nt], a, b);
    }
  }
  int half = lane >> 4;
  float* Tg = T + (size_t)g * NPAD * NPAD;
#pragma unroll
  for (int nt = 0; nt < 4; nt++)
    for (int r = 0; r < 8; r++) {
      int mm = m0 + r + half * 8;
      Tg[mm * NPAD + nb + (nt << 4) + row] = acc[nt][r];
    }
}

// D_new = S_sel^T @ T, diag & out-of-range zeroed, written back into D
// grid.x = 8 * 2, grid.y = G (gathered A loaded once, reused for 4 tiles)
__global__ void kDnew(const float* S, const int* PERM, const float* T, int k,
                      float* D) {
  int g  = blockIdx.y;
  int m0 = (blockIdx.x >> 1) << 4;
  int nb = (blockIdx.x & 1) << 6;
  const float* Sg = S + (size_t)g * NPAD * NPAD;
  const float* Tg = T + (size_t)g * NPAD * NPAD;
  const int* pg = PERM + g * NPAD;
  int lane = threadIdx.x & 31;
  int row  = lane & 15;
  int kb   = (lane >> 4) << 1;
  int am   = m0 + row;
  int pm   = (am < k) ? pg[am] : 0;
  float mmask = (am < k) ? 1.0f : 0.0f;
  v8f acc[4];
#pragma unroll
  for (int nt = 0; nt < 4; nt++) acc[nt] = vzero8();
  for (int k0 = 0; k0 < NPAD; k0 += 4) {
    v2f a;                                // S_sel^T[m][kk] = S[kk][perm[m]]
    a.x = Sg[(k0 + kb + 0) * NPAD + pm] * mmask;
    a.y = Sg[(k0 + kb + 1) * NPAD + pm] * mmask;
#pragma unroll
    for (int nt = 0; nt < 4; nt++) {
      int col = nb + (nt << 4) + row;
      v2f b;
      b.x = Tg[(k0 + kb + 0) * NPAD + col];
      b.y = Tg[(k0 + kb + 1) * NPAD + col];
      acc[nt] = wstep(acc[nt], a, b);
    }
  }
  int half = lane >> 4;
  float* Dg = D + (size_t)g * NPAD * NPAD;
#pragma unroll
  for (int nt = 0; nt < 4; nt++) {
    int col = nb + (nt << 4) + row;
    for (int r = 0; r < 8; r++) {
      int mm = m0 + r + half * 8;
      float v = (mm < k && col < k && mm != col) ? acc[nt][r] : 0.0f;
      Dg[mm * NPAD + col] = v;
    }
  }
}

// ---------------------------------------------------------------------------
// Classifier head: concat(xs) -> lin1 -> relu -> lin2 -> log_softmax
// ---------------------------------------------------------------------------
__global__ void kHead(const float* XS, const float* W1, const float* b1,
                      const float* W2, const float* b2, float* out) {
  __shared__ float z[NL * HID];
  __shared__ float t1[HID];
  __shared__ float lg[NCLS];
  __shared__ float red[2];
  int g = blockIdx.x, t = threadIdx.x;   // 64 threads
  for (int l = 0; l < NL; l++)
    z[l * HID + t] = XS[(size_t)l * G * HID + g * HID + t];
  __syncthreads();
  {
    float acc = b1[t];
    for (int kk = 0; kk < NL * HID; kk++) acc += z[kk] * W1[t * (NL * HID) + kk];
    t1[t] = acc > 0.0f ? acc : 0.0f;
  }
  __syncthreads();
  if (t < NCLS) {
    float acc = b2[t];
    for (int kk = 0; kk < HID; kk++) acc += t1[kk] * W2[t * HID + kk];
    lg[t] = acc;
  }
  __syncthreads();
  if (t == 0) {
    float mx = lg[0];
    for (int c = 1; c < NCLS; c++) mx = lg[c] > mx ? lg[c] : mx;
    float s = 0.0f;
    for (int c = 0; c < NCLS; c++) s += expf(lg[c] - mx);
    red[0] = mx;
    red[1] = logf(s);
  }
  __syncthreads();
  if (t < NCLS) out[g * NCLS + t] = lg[t] - red[0] - red[1];
}

// ---------------------------------------------------------------------------
extern "C" void kernel_launch(void* const* d_in, const int* in_sizes, int n_in,
                              void* d_out, int out_size, void* d_ws,
                              size_t ws_size, hipStream_t stream) {
  const float* x       = (const float*)d_in[0];
  const float* W_rel1  = (const float*)d_in[1];
  const float* b_rel1  = (const float*)d_in[2];
  const float* W_root1 = (const float*)d_in[3];
  const float* W_rel   = (const float*)d_in[4];
  const float* b_rel   = (const float*)d_in[5];
  const float* W_root  = (const float*)d_in[6];
  const float* pW_lin  = (const float*)d_in[7];
  const float* pb_lin  = (const float*)d_in[8];
  const float* pWa     = (const float*)d_in[9];
  const float* pWb     = (const float*)d_in[10];
  const float* pb_att  = (const float*)d_in[11];
  const float* leW1    = (const float*)d_in[12];
  const float* leb1    = (const float*)d_in[13];
  const float* leW2    = (const float*)d_in[14];
  const float* leW3    = (const float*)d_in[15];
  const float* leb3    = (const float*)d_in[16];
  const float* W_lin1  = (const float*)d_in[17];
  const float* b_lin1  = (const float*)d_in[18];
  const float* W_lin2  = (const float*)d_in[19];
  const float* b_lin2  = (const float*)d_in[20];
  const int*   ei      = (const int*)d_in[21];
  int E = in_sizes[21] / 2;

  // workspace layout (floats)
  float* base = (float*)d_ws;
  float* D    = base;                          // G*128*128
  float* DSL  = D    + G * NPAD * NPAD;
  float* S    = DSL  + G * NPAD * NPAD;
  float* TAGG = S    + G * NPAD * NPAD;        // agg (128xF) / T (128x128)
  float* HA   = TAGG + G * NPAD * FIN;
  float* HB   = HA   + G * NPAD * HID;
  float* XQ   = HB   + G * NPAD * HID;
  float* XQL  = XQ   + G * NPAD * HID;
  float* XN   = XQL  + G * NPAD * HID;
  float* RDEG = XN   + G * NPAD * HID;
  float* DEGF = RDEG + G * NPAD;
  float* XB   = DEGF + G * NPAD;
  float* QA   = XB   + G * NPAD;
  float* SAb  = QA   + G * NPAD;
  float* SBb  = SAb  + G * NPAD;
  float* SCb  = SBb  + G * NPAD;
  float* FIT  = SCb  + G * NPAD;
  float* XS   = FIT  + G * NPAD;               // NL*G*HID
  int*   PERM = (int*)(XS + NL * G * HID);     // G*128 ints

  // adjacency + initial degree
  kZero<<<256, 256, 0, stream>>>(D, G * NPAD * NPAD);
  kAdj<<<(E + 255) / 256, 256, 0, stream>>>(ei, E, D);
  kDeg<<<(G * NPAD + 255) / 256, 256, 0, stream>>>(D, RDEG);

  int n = NPAD;

  // layer 1 conv (F_IN = 128)
  kAggT<<<dim3(8 * (FIN / 64), G), 32, 0, stream>>>(D, x, RDEG, TAGG, FIN);
  kConvOut<<<dim3(8, G), 32, 0, stream>>>(TAGG, x, W_rel1, b_rel1, W_root1, HA,
                                          FIN, n);
  kMean<<<G, HID, 0, stream>>>(HA, XS, n);

  float* cur = HA;
  float* nxt = HB;
  int p = 0;
  for (int i = 0; i < NL - 1; i++) {
    kAggT<<<dim3(8 * (HID / 64), G), 32, 0, stream>>>(D, cur, RDEG, TAGG, HID);
    kConvOut<<<dim3(8, G), 32, 0, stream>>>(
        TAGG, cur, W_rel + (size_t)i * HID * HID, b_rel + i * HID,
        W_root + (size_t)i * HID * HID, nxt, HID, n);
    kMean<<<G, HID, 0, stream>>>(nxt, XS + (size_t)(i + 1) * G * HID, n);
    { float* t2 = cur; cur = nxt; nxt = t2; }

    if ((i % 2 == 0) && (i < NL - 2)) {
      int k = (9 * n + 9) / 10;   // ceil(0.9*n): 128->116->105->95->86
      kDsl<<<(G * NPAD * NPAD + 255) / 256, 256, 0, stream>>>(D, DSL);
      kDegf<<<(G * NPAD + 255) / 256, 256, 0, stream>>>(DSL, DEGF);
      kXq<<<(G * NPAD * HID + 255) / 256, 256, 0, stream>>>(DSL, cur, XQ);
      kXqLin<<<dim3(8, G), 32, 0, stream>>>(XQ, pW_lin + (size_t)p * HID * HID,
                                            pb_lin + p * HID, XQL);
      kVecs<<<(G * NPAD + 255) / 256, 256, 0, stream>>>(
          cur, XQL, pWb + p * HID, pWa + p * HID, XB, QA);
      kScore<<<(G * NPAD + 255) / 256, 256, 0, stream>>>(DSL, XB, QA,
                                                         pb_att + p, S);
      kXnew<<<dim3(8, G), 32, 0, stream>>>(S, cur, XN);
      kAB<<<(G * NPAD + 255) / 256, 256, 0, stream>>>(
          XN, leW1 + p * HID, leb1 + p, leW2 + p * HID, leW3 + p * HID,
          leb3 + p, SAb, SBb, SCb);
      kFit<<<(G * NPAD + 255) / 256, 256, 0, stream>>>(DSL, DEGF, SAb, SBb,
                                                       SCb, FIT);
      kTopK<<<1, G, 0, stream>>>(FIT, n, k, PERM);
      kXout<<<(G * NPAD * HID + 255) / 256, 256, 0, stream>>>(XN, FIT, PERM, k,
                                                              nxt);
      kT<<<dim3(16, G), 32, 0, stream>>>(DSL, S, PERM, k, TAGG);
      kDnew<<<dim3(16, G), 32, 0, stream>>>(S, PERM, TAGG, k, D);
      kDeg<<<(G * NPAD + 255) / 256, 256, 0, stream>>>(D, RDEG);  // D changed
      { float* t2 = cur; cur = nxt; nxt = t2; }
      n = k;
      p++;
    }
  }

  kHead<<<G, HID, 0, stream>>>(XS, W_lin1, b_lin1, W_lin2, b_lin2,
                               (float*)d_out);
}